// NormalizedCombinedLoss_35751307771970
// MI455X (gfx1250) — compile-verified
//
#include <hip/hip_runtime.h>

#define N_NODES  131072
#define N_EDGES  8388608
#define N_GRAPHS 64

// d_ws float layout:
// [  0.. 63] sum_ratio      per graph
// [ 64..127] sum_ratio2     per graph
// [128..191] sum_overlap    per graph
// [192..255] edge_cnt       per graph
// [256..319] node_cnt       per graph
// [320..383] graph_stress   per graph
#define WS_SUMR    0
#define WS_SUMR2   64
#define WS_OVL     128
#define WS_ECNT    192
#define WS_NCNT    256
#define WS_STRESS  320
#define WS_FLOATS  384

typedef __attribute__((ext_vector_type(2))) float v2f;
typedef __attribute__((ext_vector_type(8))) float v8f;

// Hardware v_rcp_f32 / v_sqrt_f32 (~1 ulp): per-edge math is tolerance-safe
// at 1e-7 relative, and avoids the IEEE div_scale/div_fmas/div_fixup chains.
__device__ __forceinline__ float fast_rcp(float x)  { return __builtin_amdgcn_rcpf(x); }
__device__ __forceinline__ float fast_sqrt(float x) { return __builtin_amdgcn_sqrtf(x); }

// ---------------------------------------------------------------------------
// Node counting: graph_sizes[g] = #nodes with batch_idx==g
// ---------------------------------------------------------------------------
__global__ __launch_bounds__(256) void nodecnt_kernel(
    const int* __restrict__ batch, float* __restrict__ ws)
{
    __shared__ float bins[2 * N_GRAPHS];
    const int tid = threadIdx.x;
    if (tid < 2 * N_GRAPHS) bins[tid] = 0.0f;
    __syncthreads();
    const int rep = (tid & 1) * N_GRAPHS;
    // 64 blocks * 256 threads * 8 nodes = 131072
    const int i0 = blockIdx.x * 2048 + tid;
#pragma unroll
    for (int k = 0; k < 8; ++k) {
        const int g = batch[i0 + k * 256];
        atomicAdd(&bins[rep + g], 1.0f);
    }
    __syncthreads();
    if (tid < N_GRAPHS) atomicAdd(&ws[WS_NCNT + tid], bins[tid] + bins[N_GRAPHS + tid]);
}

// ---------------------------------------------------------------------------
// Pass 1 over edges: per-graph sum(ratio), sum(ratio^2), sum(norm_overlap),
// edge counts. 2048 blocks, each handles 4096 edges as 4 iters of
// 256 threads x int4/float4 (coalesced b128 loads). LDS bins replicated 2x
// by lane parity to cut ds_add_f32 same-address serialization.
// ---------------------------------------------------------------------------
__global__ __launch_bounds__(256) void pass1_kernel(
    const int*    __restrict__ idx0,
    const int*    __restrict__ idx1,
    const float*  __restrict__ attr,
    const float2* __restrict__ pos,
    const float2* __restrict__ sizes,
    const int*    __restrict__ batch,
    float*        __restrict__ ws)
{
    __shared__ float bins[2 * 4 * N_GRAPHS];   // 2 replicas of [r, r2, ovl, cnt] x 64
    const int tid = threadIdx.x;
    bins[tid]       = 0.0f;
    bins[tid + 256] = 0.0f;
    __syncthreads();
    const int rep = (tid & 1) * 256;

    const int4*   v0 = (const int4*)idx0;
    const int4*   v1 = (const int4*)idx1;
    const float4* va = (const float4*)attr;
    const int base4 = blockIdx.x * 1024;    // in int4 units

    for (int it = 0; it < 4; ++it) {
        const int q = base4 + it * 256 + tid;
        if (it < 3) {                        // stream-ahead prefetch (in-bounds)
            __builtin_prefetch(&v0[q + 256], 0, 1);
            __builtin_prefetch(&v1[q + 256], 0, 1);
            __builtin_prefetch(&va[q + 256], 0, 1);
        }
        const int4   s4 = v0[q];
        const int4   t4 = v1[q];
        const float4 d4 = va[q];
        const int   si[4] = {s4.x, s4.y, s4.z, s4.w};
        const int   ti[4] = {t4.x, t4.y, t4.z, t4.w};
        const float di[4] = {d4.x, d4.y, d4.z, d4.w};
#pragma unroll
        for (int k = 0; k < 4; ++k) {
            const int s = si[k], t = ti[k];
            const float d = di[k];
            const float2 ps = pos[s];
            const float2 pt = pos[t];
            const int g = batch[s];
            const float dx = ps.x - pt.x;
            const float dy = ps.y - pt.y;
            const float eu = fast_sqrt(dx * dx + dy * dy);
            const float r = eu * fast_rcp(d);
            atomicAdd(&bins[rep + WS_SUMR  + g], r);
            atomicAdd(&bins[rep + WS_SUMR2 + g], r * r);
            const float2 ss = sizes[s];
            const float2 st = sizes[t];
            const float ox = fmaxf((ss.x + st.x) * 0.5f - fabsf(dx), 0.0f);
            const float oy = fmaxf((ss.y + st.y) * 0.5f - fabsf(dy), 0.0f);
            const float ov = (ox * oy) * fast_rcp(ss.x + ss.y + st.x + st.y);
            atomicAdd(&bins[rep + WS_OVL  + g], ov);
            atomicAdd(&bins[rep + WS_ECNT + g], 1.0f);
        }
    }
    __syncthreads();
    atomicAdd(&ws[tid], bins[tid] + bins[tid + 256]);   // ws[0..255] matches bins
}

// ---------------------------------------------------------------------------
// Pass 2 over edges: stress with per-endpoint graph scales.
// scale[g] = sum_r2/sum_r ; scaled_pos = pos * (sum_r/sum_r2) = pos * inv[g]
// Per-graph scale kept as exact IEEE division (64 ops only).
// ---------------------------------------------------------------------------
__global__ __launch_bounds__(256) void pass2_kernel(
    const int*    __restrict__ idx0,
    const int*    __restrict__ idx1,
    const float*  __restrict__ attr,
    const float2* __restrict__ pos,
    const int*    __restrict__ batch,
    float*        __restrict__ ws)
{
    __shared__ float inv[N_GRAPHS];
    __shared__ float sbin[2 * N_GRAPHS];
    const int tid = threadIdx.x;
    if (tid < N_GRAPHS) {
        const float sr  = ws[WS_SUMR  + tid];
        const float sr2 = ws[WS_SUMR2 + tid];
        inv[tid]  = sr / sr2;    // 1/scale (exact)
        sbin[tid]            = 0.0f;
        sbin[N_GRAPHS + tid] = 0.0f;
    }
    __syncthreads();
    const int rep = (tid & 1) * N_GRAPHS;

    const int4*   v0 = (const int4*)idx0;
    const int4*   v1 = (const int4*)idx1;
    const float4* va = (const float4*)attr;
    const int base4 = blockIdx.x * 1024;

    for (int it = 0; it < 4; ++it) {
        const int q = base4 + it * 256 + tid;
        if (it < 3) {
            __builtin_prefetch(&v0[q + 256], 0, 1);
            __builtin_prefetch(&v1[q + 256], 0, 1);
            __builtin_prefetch(&va[q + 256], 0, 1);
        }
        const int4   s4 = v0[q];
        const int4   t4 = v1[q];
        const float4 d4 = va[q];
        const int   si[4] = {s4.x, s4.y, s4.z, s4.w};
        const int   ti[4] = {t4.x, t4.y, t4.z, t4.w};
        const float di[4] = {d4.x, d4.y, d4.z, d4.w};
#pragma unroll
        for (int k = 0; k < 4; ++k) {
            const int s = si[k], t = ti[k];
            const float d = di[k];
            const float2 ps = pos[s];
            const float2 pt = pos[t];
            const int gs = batch[s];
            const int gt = batch[t];
            const float fs = inv[gs];
            const float ft = inv[gt];
            const float dx = ps.x * fs - pt.x * ft;
            const float dy = ps.y * fs - pt.y * ft;
            const float eu2 = fast_sqrt(dx * dx + dy * dy);
            const float u = (eu2 - d) * fast_rcp(d);
            atomicAdd(&sbin[rep + gs], u * u);
        }
    }
    __syncthreads();
    if (tid < N_GRAPHS)
        atomicAdd(&ws[WS_STRESS + tid], sbin[tid] + sbin[N_GRAPHS + tid]);
}

// ---------------------------------------------------------------------------
// Final combine: combined[g] = stress/(ncnt^2) + ovl/max(ecnt,1);
// mean over 64 graphs via V_WMMA_F32_16X16X4_F32 (exact f32 matrix reduce:
// A = 16x4 of combined values, B = ones(4x16), row-sums land in D column 0).
// One wave32, EXEC all-ones at the WMMA. Per-graph divides stay exact IEEE.
// ---------------------------------------------------------------------------
__global__ __launch_bounds__(32) void final_kernel(
    const float* __restrict__ ws, float* __restrict__ out)
{
    __shared__ float comb[N_GRAPHS];
    __shared__ float part[32];
    const int tid = threadIdx.x;
#pragma unroll
    for (int g = tid; g < N_GRAPHS; g += 32) {
        const float stress = ws[WS_STRESS + g];
        const float nc     = ws[WS_NCNT   + g];
        const float ov     = ws[WS_OVL    + g];
        const float ec     = ws[WS_ECNT   + g];
        comb[g] = stress / (nc * nc) + ov / fmaxf(ec, 1.0f);
    }
    __syncthreads();

    // A-matrix 16x4 f32 layout: lanes 0-15 hold row M=lane (K=0,1 in the two
    // VGPRs); lanes 16-31 hold the same rows with K=2,3.
    const int m  = tid & 15;
    const int kh = tid >> 4;
    v2f a;
    a.x = comb[(2 * kh + 0) * 16 + m];
    a.y = comb[(2 * kh + 1) * 16 + m];
    v2f b;
    b.x = 1.0f;
    b.y = 1.0f;
    v8f c = {};
    v8f dm = __builtin_amdgcn_wmma_f32_16x16x4_f32(
        /*neg_a=*/false, a, /*neg_b=*/false, b,
        /*c_mod=*/(short)0, c, /*reuse_a=*/false, /*reuse_b=*/false);
    // Lane n holds D[m][n] for m = 0..7 (+8 in lanes 16-31); with B==1 every
    // column equals the A row-sums, so lane0 + lane16 partials = total sum.
    float local = dm[0] + dm[1] + dm[2] + dm[3] + dm[4] + dm[5] + dm[6] + dm[7];
    part[tid] = local;
    __syncthreads();
    if (tid == 0) out[0] = (part[0] + part[16]) * (1.0f / (float)N_GRAPHS);
}

// ---------------------------------------------------------------------------
extern "C" void kernel_launch(void* const* d_in, const int* in_sizes, int n_in,
                              void* d_out, int out_size, void* d_ws, size_t ws_size,
                              hipStream_t stream) {
    (void)in_sizes; (void)n_in; (void)out_size; (void)ws_size;
    const float2* node_pos   = (const float2*)d_in[0];
    const float2* node_sizes = (const float2*)d_in[1];
    const float*  attr       = (const float*)d_in[2];
    const int*    eidx       = (const int*)d_in[3];
    const int*    batch      = (const int*)d_in[4];
    const int*    idx0       = eidx;
    const int*    idx1       = eidx + N_EDGES;
    float*        ws         = (float*)d_ws;
    float*        out        = (float*)d_out;

    hipMemsetAsync(d_ws, 0, WS_FLOATS * sizeof(float), stream);
    nodecnt_kernel<<<N_NODES / 2048, 256, 0, stream>>>(batch, ws);
    pass1_kernel<<<N_EDGES / 4096, 256, 0, stream>>>(
        idx0, idx1, attr, node_pos, node_sizes, batch, ws);
    pass2_kernel<<<N_EDGES / 4096, 256, 0, stream>>>(
        idx0, idx1, attr, node_pos, batch, ws);
    final_kernel<<<1, 32, 0, stream>>>(ws, out);
}